// DistanceMatrix_3049426780789
// MI455X (gfx1250) — compile-verified
//
#include <hip/hip_runtime.h>

#define K_DIM 64
#define EPS_F 1e-8f

typedef __attribute__((ext_vector_type(2))) float v2f;
typedef __attribute__((ext_vector_type(8))) float v8f;

// ---------------------------------------------------------------------------
// Kernel 1: row-normalize A and B into workspace.
// One wave32 per row; each lane owns a float2 (64 elements / 32 lanes).
// ---------------------------------------------------------------------------
__global__ __launch_bounds__(256) void cosine_normalize_rows(
    const float* __restrict__ A, const float* __restrict__ B,
    float* __restrict__ An, float* __restrict__ Bn, int nA, int nB) {
  int row  = blockIdx.x * 8 + (threadIdx.x >> 5);   // 8 waves per block
  int lane = threadIdx.x & 31;

  const float* src;
  float* dst;
  int r;
  if (row < nA) {
    src = A; dst = An; r = row;
  } else {
    r = row - nA;
    if (r >= nB) return;
    src = B; dst = Bn;
  }

  const float2* srow = (const float2*)(src + (size_t)r * K_DIM);
  float2 v = srow[lane];
  float s = v.x * v.x + v.y * v.y;
#pragma unroll
  for (int off = 16; off > 0; off >>= 1) s += __shfl_xor(s, off, 32);

  float scale = 1.0f / fmaxf(sqrtf(s), EPS_F);
  float2 o;
  o.x = v.x * scale;
  o.y = v.y * scale;
  ((float2*)(dst + (size_t)r * K_DIM))[lane] = o;
}

// ---------------------------------------------------------------------------
// Kernel 2: Out[16x16 tile] = An_tile(16x64) * Bn_tile(16x64)^T via
// 16 chained V_WMMA_F32_16X16X4_F32. One wave per output tile.
//
// f32 A-fragment (16x4) per ISA layout: lane l<16 -> row M=l, K=k0..k0+1;
// lanes 16..31 -> row M=l-16, K=k0+2..k0+3.  The B (4x16) fragment for
// B^T follows the mirrored layout, so both sides are one aligned float2
// load per lane per k-step.
// ---------------------------------------------------------------------------
__global__ __launch_bounds__(256) void cosine_gemm_wmma(
    const float* __restrict__ An, const float* __restrict__ Bn,
    float* __restrict__ Out, int M, int N) {
  int lane = threadIdx.x & 31;
  int wave = threadIdx.x >> 5;

  int tilesN = N >> 4;
  int tile = blockIdx.x * 8 + wave;
  int tM = tile / tilesN;          // wave-uniform
  int tN = tile - tM * tilesN;
  if (tM >= (M >> 4)) return;      // wave-uniform guard: EXEC stays all-ones

  int half = lane >> 4;            // 0: K+0..1, 1: K+2..3
  int lr   = lane & 15;

  const float* aP = An + (size_t)(tM * 16 + lr) * K_DIM + 2 * half;
  const float* bP = Bn + (size_t)(tN * 16 + lr) * K_DIM + 2 * half;

  v8f c = {};
#pragma unroll
  for (int k0 = 0; k0 < K_DIM; k0 += 4) {
    v2f a = *(const v2f*)(aP + k0);
    v2f b = *(const v2f*)(bP + k0);
    // (neg_a, A, neg_b, B, c_mod, C, reuse_a, reuse_b)
    c = __builtin_amdgcn_wmma_f32_16x16x4_f32(false, a, false, b,
                                              (short)0, c, false, false);
  }

  // C/D layout: VGPR v, lane l -> row (v + 8*(l>=16)), col (l & 15).
  float* orow = Out + (size_t)(tM * 16 + half * 8) * N + (tN * 16 + lr);
#pragma unroll
  for (int v = 0; v < 8; ++v) orow[(size_t)v * N] = c[v];
}

// ---------------------------------------------------------------------------
extern "C" void kernel_launch(void* const* d_in, const int* in_sizes, int n_in,
                              void* d_out, int out_size, void* d_ws, size_t ws_size,
                              hipStream_t stream) {
  const float* A = (const float*)d_in[0];
  const float* B = (const float*)d_in[1];
  int nA = in_sizes[0] / K_DIM;    // 8192
  int nB = in_sizes[1] / K_DIM;    // 8192

  float* An = (float*)d_ws;                       // nA*64 floats
  float* Bn = An + (size_t)nA * K_DIM;            // nB*64 floats (4 MB total)
  float* Out = (float*)d_out;

  int rows = nA + nB;
  cosine_normalize_rows<<<(rows + 7) / 8, 256, 0, stream>>>(A, B, An, Bn, nA, nB);

  int tiles = (nA >> 4) * (nB >> 4);              // 512*512
  cosine_gemm_wmma<<<(tiles + 7) / 8, 256, 0, stream>>>(An, Bn, Out, nA, nB);
}